// SourceBiasSeq_38328288149532
// MI455X (gfx1250) — compile-verified
//
#include <hip/hip_runtime.h>
#include <hip/hip_bf16.h>
#include <cmath>

// Problem constants (from reference): B=64, S=16, D=1024, E=64, N=B*S=1024
#define DD   1024
#define EE   64
#define NTOK 1024

typedef __attribute__((ext_vector_type(2))) float v2f;
typedef __attribute__((ext_vector_type(8))) float v8f;

// ---------------------------------------------------------------------------
// Phase 1: build per-expert token lists.
// counts[e] = number of tokens routed to expert e
// lists[e*NTOK + i] = token index (order nondeterministic; output is not)
// ---------------------------------------------------------------------------
__global__ void build_lists_kernel(const int* __restrict__ urls,
                                   int* __restrict__ counts,
                                   int* __restrict__ lists) {
    int n = blockIdx.x * blockDim.x + threadIdx.x;
    if (n < NTOK) {
        int e = urls[n];
        int p = atomicAdd(&counts[e], 1);
        lists[e * NTOK + p] = n;
    }
}

// ---------------------------------------------------------------------------
// Phase 2: routed GEMM via V_WMMA_F32_16X16X4_F32.
// Grid: (EE, DD/64). Block: 128 threads = 4 waves; wave w owns a 16-column
// tile [blockIdx.y*64 + 16*w, +16). Loop over token tiles of 16 rows.
// A (16x4 f32): lane L holds row M=L%16; VGPR0={K0|K2}, VGPR1={K1|K3}
//               split across lane halves.
// B (4x16 f32): lane L holds col N=L%16; same K striping over lane halves.
// C/D (16x16 f32, 8 VGPRs): VGPR r holds M = r + 8*(L>=16), N = L%16.
// ---------------------------------------------------------------------------
__global__ void __launch_bounds__(128)
moe_wmma_kernel(const float* __restrict__ x,     // [NTOK, DD]
                const float* __restrict__ trans, // [EE, DD, DD]
                const float* __restrict__ bias,  // [EE, DD]
                const int* __restrict__ counts,
                const int* __restrict__ lists,
                float* __restrict__ out) {       // [NTOK, DD]
    const int e   = blockIdx.x;
    const int cnt = counts[e];
    if (cnt == 0) return;

    const int lane = threadIdx.x & 31;
    const int wave = threadIdx.x >> 5;
    const int hi   = lane >> 4;                       // 0: lanes 0-15, 1: 16-31
    const int nloc = lane & 15;
    const int col  = blockIdx.y * 64 + wave * 16 + nloc;

    const float* __restrict__ W = trans + (size_t)e * DD * DD;
    const float  bv = bias[e * DD + col];

    __shared__ float xs[16 * DD];   // 64 KB: 16 token rows of x
    __shared__ int   tok[16];

    for (int tile = 0; tile < cnt; tile += 16) {
        __syncthreads();            // protect previous tile's LDS use
        if (threadIdx.x < 16) {
            int idx = tile + threadIdx.x;
            tok[threadIdx.x] = lists[e * NTOK + (idx < cnt ? idx : tile)];
        }
        __syncthreads();

        // Cooperative, coalesced float4 stage of 16 rows of x into LDS.
        for (int i = threadIdx.x; i < 16 * (DD / 4); i += 128) {
            int r  = i >> 8;          // DD/4 = 256 float4 per row
            int c4 = i & 255;
            float4 v = reinterpret_cast<const float4*>(x + (size_t)tok[r] * DD)[c4];
            reinterpret_cast<float4*>(xs + r * DD)[c4] = v;
        }
        __syncthreads();

        v8f acc = {};
        const float* __restrict__ arow = xs + nloc * DD;
        #pragma unroll 4
        for (int k = 0; k < DD; k += 4) {
            const int kk = k + 2 * hi;
            v2f a, b;
            a.x = arow[kk];
            a.y = arow[kk + 1];
            b.x = W[(size_t)kk * DD + col];
            b.y = W[(size_t)(kk + 1) * DD + col];
            // D = A(16x4) * B(4x16) + C ; emits v_wmma_f32_16x16x4_f32
            acc = __builtin_amdgcn_wmma_f32_16x16x4_f32(
                /*neg_a=*/false, a, /*neg_b=*/false, b,
                /*c_mod=*/(short)0, acc, /*reuse_a=*/false, /*reuse_b=*/false);
        }

        // Epilogue: bias + tanh, masked store of valid rows.
        #pragma unroll
        for (int r = 0; r < 8; ++r) {
            int m  = r + 8 * hi;
            int gm = tile + m;
            if (gm < cnt) {
                out[(size_t)tok[m] * DD + col] = tanhf(acc[r] + bv);
            }
        }
    }
}

// ---------------------------------------------------------------------------
// Launch: inputs per setup_inputs() order: x[f32], urls[i32], trans[f32],
// bias[f32]. Output f32 [B,S,D]. Workspace: counts (256 B) + lists (256 KB).
// ---------------------------------------------------------------------------
extern "C" void kernel_launch(void* const* d_in, const int* in_sizes, int n_in,
                              void* d_out, int out_size, void* d_ws, size_t ws_size,
                              hipStream_t stream) {
    const float* x     = (const float*)d_in[0];
    const int*   urls  = (const int*)d_in[1];
    const float* trans = (const float*)d_in[2];
    const float* bias  = (const float*)d_in[3];
    float*       out   = (float*)d_out;

    int* counts = (int*)d_ws;
    int* lists  = counts + EE;   // 256 B in, 256 KB long

    hipMemsetAsync(counts, 0, EE * sizeof(int), stream);
    build_lists_kernel<<<(NTOK + 255) / 256, 256, 0, stream>>>(urls, counts, lists);

    dim3 grid(EE, DD / 64);
    moe_wmma_kernel<<<grid, 128, 0, stream>>>(x, trans, bias, counts, lists, out);
}